// WorldGraphEncoder_13924283973767
// MI455X (gfx1250) — compile-verified
//
#include <hip/hip_runtime.h>
#include <cstdint>

#define LEAKY_ALPHA 0.2f
#define SEG_EPS 1e-8f
#define LN_EPS 1e-5f

typedef float v2f __attribute__((ext_vector_type(2)));
typedef float v8f __attribute__((ext_vector_type(8)));

static __device__ __forceinline__ float wave_sum32(float v) {
#pragma unroll
  for (int m = 16; m >= 1; m >>= 1) v += __shfl_xor(v, m, 32);
  return v;
}

static __device__ __forceinline__ void atomic_add_f32(float* p, float v) {
  // hardware global_atomic_add_f32 (no CAS loop)
  unsafeAtomicAdd(p, v);
}

// ---------------------------------------------------------------------------
// Zero accumulators (n multiple of 4)
// ---------------------------------------------------------------------------
__global__ void zero_kernel(float* __restrict__ p, long long n4) {
  long long i = ((long long)blockIdx.x * blockDim.x + threadIdx.x);
  if (i < n4) {
    float4 z = make_float4(0.f, 0.f, 0.f, 0.f);
    *(float4*)(p + i * 4) = z;
  }
}

// ---------------------------------------------------------------------------
// Node feature build: x = [cat_emb(32) | spatialMLP(16) | base(16)]  -> [N,64]
// ---------------------------------------------------------------------------
__global__ void build_features_kernel(const int* __restrict__ categories,
                                      const float* __restrict__ coords,
                                      const float* __restrict__ cat_emb,
                                      const float* __restrict__ spW1,
                                      const float* __restrict__ spb1,
                                      const float* __restrict__ spW2,
                                      const float* __restrict__ spb2,
                                      const float* __restrict__ base_emb,
                                      float* __restrict__ X, int N) {
  int n = blockIdx.x * blockDim.x + threadIdx.x;
  if (n >= N) return;
  int cat = categories[n];
  float cx = coords[(size_t)n * 2 + 0];
  float cy = coords[(size_t)n * 2 + 1];
  float hb[16];
#pragma unroll
  for (int j = 0; j < 16; j++) {
    float v = cx * spW1[j] + cy * spW1[16 + j] + spb1[j];
    hb[j] = v > 0.f ? v : 0.f;
  }
  float* xr = X + (size_t)n * 64;
  const float* ce = cat_emb + (size_t)cat * 32;
#pragma unroll
  for (int j = 0; j < 32; j++) xr[j] = ce[j];
#pragma unroll
  for (int j = 0; j < 16; j++) {
    float s = spb2[j];
#pragma unroll
    for (int k = 0; k < 16; k++) s += hb[k] * spW2[k * 16 + j];
    xr[32 + j] = s;
  }
  const float* be = base_emb + (size_t)n * 16;
#pragma unroll
  for (int j = 0; j < 16; j++) xr[48 + j] = be[j];
}

// ---------------------------------------------------------------------------
// f32 WMMA GEMM:  H[N, 4*CPH] = X[N, K] @ W   (W = [4 heads][K][CPH])
// One wave computes one 16x16 output tile with V_WMMA_F32_16X16X4_F32.
// Operand layouts per CDNA5 ISA 7.12.2:
//   A: lane = M (dup halves), VGPR0/1 = K + {0,1} with K-base 0 (lanes<16) / 2
//   B: lane = N (dup halves), VGPR0/1 = K + {0,1} same half split
//   D: lane = N, VGPR v = M row (v, or 8+v for lanes>=16)
// ---------------------------------------------------------------------------
template <int K, int CPH>
__global__ void gemm_wmma_kernel(const float* __restrict__ X,
                                 const float* __restrict__ W,
                                 float* __restrict__ H, int nRows) {
  constexpr int NCOLT = (4 * CPH) / 16;   // output col tiles
  constexpr int LDH = 4 * CPH;            // output row stride
  constexpr int HSTRIDE = K * CPH;        // per-head weight block
  const int lane = threadIdx.x & 31;
  const int wave = blockIdx.x * (blockDim.x >> 5) + (threadIdx.x >> 5);
  const int nRowTiles = (nRows + 15) >> 4;
  if (wave >= nRowTiles * NCOLT) return;  // uniform per wave: EXEC stays full
  const int rowTile = wave / NCOLT;
  const int colTile = wave % NCOLT;
  const int m = lane & 15;
  const int koff = (lane >> 4) << 1;      // 0 or 2
  int r = rowTile * 16 + m;
  if (r >= nRows) r = nRows - 1;          // clamp loads; stores guarded below
  const int c = colTile * 16 + m;
  const int head = c / CPH;
  const int cc = c % CPH;
  const float* xrow = X + (size_t)r * K + koff;
  const float* wcol = W + (size_t)head * HSTRIDE + (size_t)koff * CPH + cc;
  v8f acc = {};
#pragma unroll
  for (int k0 = 0; k0 < K; k0 += 4) {
    v2f a = *(const v2f*)(xrow + k0);     // b64 load, contiguous K pair
    v2f b;
    const float* wp = wcol + (size_t)k0 * CPH;
    b.x = wp[0];
    b.y = wp[CPH];
    acc = __builtin_amdgcn_wmma_f32_16x16x4_f32(false, a, false, b, (short)0,
                                                acc, false, false);
  }
  const int outCol = colTile * 16 + m;
  const int mBase = rowTile * 16 + ((lane >> 4) << 3);
  float* hp = H + (size_t)mBase * LDH + outCol;
#pragma unroll
  for (int v = 0; v < 8; v++) {
    if (mBase + v < nRows) hp[(size_t)v * LDH] = acc[v];
  }
}

// ---------------------------------------------------------------------------
// Per-node per-head attention scores: s_src = h.a[:F], s_tgt = h.a[F:]
// stored [N,4] for float4 edge gathers
// ---------------------------------------------------------------------------
__global__ void scores_kernel(const float* __restrict__ H,
                              const float* __restrict__ a,
                              float* __restrict__ ssrc, float* __restrict__ stgt,
                              int N, int F) {
  int t = blockIdx.x * blockDim.x + threadIdx.x;
  if (t >= N * 4) return;
  int n = t >> 2, h = t & 3;
  const float* hp = H + (size_t)n * (4 * F) + (size_t)h * F;
  const float* as = a + (size_t)h * (2 * F);
  const float* at = as + F;
  float s0 = 0.f, s1 = 0.f;
  for (int f = 0; f < F; f++) {
    float v = hp[f];
    s0 += v * as[f];
    s1 += v * at[f];
  }
  ssrc[t] = s0;
  stgt[t] = s1;
}

// ---------------------------------------------------------------------------
// Edge pass A: e = leaky(s_src[src]+s_tgt[tgt])*ew; expe stored; denom scatter
// ---------------------------------------------------------------------------
__global__ void edge_attn_kernel(const int* __restrict__ eidx,
                                 const float* __restrict__ ew,
                                 const float* __restrict__ ssrc,
                                 const float* __restrict__ stgt,
                                 float* __restrict__ expe,
                                 float* __restrict__ denom, int E) {
  int e = blockIdx.x * blockDim.x + threadIdx.x;
  if (e >= E) return;
  int s = eidx[e];
  int t = eidx[E + e];
  float w = ew[e];
  float4 a4 = *(const float4*)(ssrc + (size_t)s * 4);
  float4 b4 = *(const float4*)(stgt + (size_t)t * 4);
  float av[4] = {a4.x, a4.y, a4.z, a4.w};
  float bv[4] = {b4.x, b4.y, b4.z, b4.w};
  float xv[4];
#pragma unroll
  for (int h = 0; h < 4; h++) {
    float v = av[h] + bv[h];
    v = v > 0.f ? v : LEAKY_ALPHA * v;
    v *= w;
    float ex = expf(v);
    xv[h] = ex;
    atomic_add_f32(&denom[(size_t)t * 4 + h], ex);
  }
  *(float4*)(expe + (size_t)e * 4) = make_float4(xv[0], xv[1], xv[2], xv[3]);
}

// ---------------------------------------------------------------------------
// Edge pass B: h'[tgt] += alpha * h[src].  F = per-head feats; Ftot = 4F.
// One thread = 4 contiguous feats of one edge -> one wave = 128 feats
// (layer1: exactly one edge per wave), coalesced B128 gathers + same-line
// f32 atomics.
// ---------------------------------------------------------------------------
template <int F>
__global__ void edge_scatter_kernel(const int* __restrict__ eidx,
                                    const float* __restrict__ expe,
                                    const float* __restrict__ denom,
                                    const float* __restrict__ H,
                                    float* __restrict__ hacc, int E) {
  constexpr int FT = 4 * F;
  long long tid = (long long)blockIdx.x * blockDim.x + threadIdx.x;
  if (tid >= (long long)E * F) return;   // F chunks of 4 per edge
  int e = (int)(tid / F);
  int q = (int)(tid % F);
  int f = q << 2;
  int h = f / F;                          // head of this 4-chunk
  int s = eidx[e];
  int t = eidx[E + e];
  float al = expe[(size_t)e * 4 + h] / (denom[(size_t)t * 4 + h] + SEG_EPS);
  float4 hv = *(const float4*)(H + (size_t)s * FT + f);
  float* dst = hacc + (size_t)t * FT + f;
  atomic_add_f32(dst + 0, hv.x * al);
  atomic_add_f32(dst + 1, hv.y * al);
  atomic_add_f32(dst + 2, hv.z * al);
  atomic_add_f32(dst + 3, hv.w * al);
}

// ---------------------------------------------------------------------------
// Layer1 epilogue: ELU then LayerNorm(128) -> X2.  One wave per node.
// ---------------------------------------------------------------------------
__global__ void elu_ln_kernel(const float* __restrict__ hacc,
                              const float* __restrict__ g,
                              const float* __restrict__ b,
                              float* __restrict__ X2, int N) {
  int lane = threadIdx.x & 31;
  int node = blockIdx.x * (blockDim.x >> 5) + (threadIdx.x >> 5);
  if (node >= N) return;
  const float* hp = hacc + (size_t)node * 128 + lane * 4;
  float v[4];
#pragma unroll
  for (int i = 0; i < 4; i++) {
    float x = hp[i];
    v[i] = x > 0.f ? x : expm1f(x);
  }
  float mean = wave_sum32(v[0] + v[1] + v[2] + v[3]) * (1.0f / 128.0f);
  float d = 0.f;
#pragma unroll
  for (int i = 0; i < 4; i++) {
    float t = v[i] - mean;
    d += t * t;
  }
  d = wave_sum32(d);
  float rstd = rsqrtf(d * (1.0f / 128.0f) + LN_EPS);
  float* xo = X2 + (size_t)node * 128 + lane * 4;
#pragma unroll
  for (int i = 0; i < 4; i++) {
    int f = lane * 4 + i;
    xo[i] = (v[i] - mean) * rstd * g[f] + b[f];
  }
}

// ---------------------------------------------------------------------------
// Layer2 epilogue: mean over heads of ELU, LayerNorm(128), + residual X2.
// ---------------------------------------------------------------------------
__global__ void final_kernel(const float* __restrict__ hacc,
                             const float* __restrict__ X2,
                             const float* __restrict__ g,
                             const float* __restrict__ b,
                             float* __restrict__ out, int N) {
  int lane = threadIdx.x & 31;
  int node = blockIdx.x * (blockDim.x >> 5) + (threadIdx.x >> 5);
  if (node >= N) return;
  const float* hp = hacc + (size_t)node * 512;
  float v[4];
#pragma unroll
  for (int i = 0; i < 4; i++) {
    int o = lane * 4 + i;
    float acc = 0.f;
#pragma unroll
    for (int h = 0; h < 4; h++) {
      float x = hp[h * 128 + o];
      acc += x > 0.f ? x : expm1f(x);
    }
    v[i] = acc * 0.25f;
  }
  float mean = wave_sum32(v[0] + v[1] + v[2] + v[3]) * (1.0f / 128.0f);
  float d = 0.f;
#pragma unroll
  for (int i = 0; i < 4; i++) {
    float t = v[i] - mean;
    d += t * t;
  }
  d = wave_sum32(d);
  float rstd = rsqrtf(d * (1.0f / 128.0f) + LN_EPS);
#pragma unroll
  for (int i = 0; i < 4; i++) {
    int o = lane * 4 + i;
    out[(size_t)node * 128 + o] =
        (v[i] - mean) * rstd * g[o] + b[o] + X2[(size_t)node * 128 + o];
  }
}

// ---------------------------------------------------------------------------
extern "C" void kernel_launch(void* const* d_in, const int* in_sizes, int n_in,
                              void* d_out, int out_size, void* d_ws,
                              size_t ws_size, hipStream_t stream) {
  const int* categories = (const int*)d_in[0];
  const float* coords = (const float*)d_in[1];
  const int* eidx = (const int*)d_in[2];
  const float* ew = (const float*)d_in[3];
  const float* cat_emb = (const float*)d_in[4];
  const float* spW1 = (const float*)d_in[5];
  const float* spb1 = (const float*)d_in[6];
  const float* spW2 = (const float*)d_in[7];
  const float* spb2 = (const float*)d_in[8];
  const float* base_emb = (const float*)d_in[9];
  const float* W1 = (const float*)d_in[10];
  const float* a1 = (const float*)d_in[11];
  const float* ln1g = (const float*)d_in[12];
  const float* ln1b = (const float*)d_in[13];
  const float* W2 = (const float*)d_in[14];
  const float* a2 = (const float*)d_in[15];
  const float* ln2g = (const float*)d_in[16];
  const float* ln2b = (const float*)d_in[17];

  const int N = in_sizes[0];
  const int E = in_sizes[3];
  float* out = (float*)d_out;

  // workspace layout (floats)
  float* ws = (float*)d_ws;
  size_t o = 0;
  float* X1 = ws + o;    o += (size_t)N * 64;
  float* H1 = ws + o;    o += (size_t)N * 128;
  float* ssrc1 = ws + o; o += (size_t)N * 4;
  float* stgt1 = ws + o; o += (size_t)N * 4;
  float* ssrc2 = ws + o; o += (size_t)N * 4;
  float* stgt2 = ws + o; o += (size_t)N * 4;
  float* expe = ws + o;  o += (size_t)E * 4;   // reused by both layers
  float* X2 = ws + o;    o += (size_t)N * 128;
  float* H2 = ws + o;    o += (size_t)N * 512;
  // zeroed accumulator block (contiguous)
  float* denom1 = ws + o; o += (size_t)N * 4;
  float* denom2 = ws + o; o += (size_t)N * 4;
  float* hacc1 = ws + o;  o += (size_t)N * 128;
  float* hacc2 = ws + o;  o += (size_t)N * 512;

  const int TB = 256;
  // 0) zero accumulators (648 floats per node, multiple of 4)
  long long z4 = (long long)N * 648 / 4;
  zero_kernel<<<(int)((z4 + TB - 1) / TB), TB, 0, stream>>>(denom1, z4);

  // 1) node features
  build_features_kernel<<<(N + TB - 1) / TB, TB, 0, stream>>>(
      categories, coords, cat_emb, spW1, spb1, spW2, spb2, base_emb, X1, N);

  // 2) layer-1 GEMM via f32 WMMA: [N,64] x [64,128]
  {
    int rowTiles = (N + 15) / 16;
    int waves = rowTiles * 8;  // 8 col tiles
    gemm_wmma_kernel<64, 32><<<(waves + 7) / 8, TB, 0, stream>>>(X1, W1, H1, N);
  }
  // 3) layer-1 scores
  scores_kernel<<<(N * 4 + TB - 1) / TB, TB, 0, stream>>>(H1, a1, ssrc1, stgt1, N, 32);
  // 4) layer-1 edge softmax numerators + denominators
  edge_attn_kernel<<<(E + TB - 1) / TB, TB, 0, stream>>>(eidx, ew, ssrc1, stgt1,
                                                         expe, denom1, E);
  // 5) layer-1 weighted scatter (wave per edge: 32 thr x 4 feats = 128)
  {
    long long thr = (long long)E * 32;
    edge_scatter_kernel<32><<<(int)((thr + TB - 1) / TB), TB, 0, stream>>>(
        eidx, expe, denom1, H1, hacc1, E);
  }
  // 6) ELU + LayerNorm -> X2 (wave per node)
  elu_ln_kernel<<<(N + 7) / 8, TB, 0, stream>>>(hacc1, ln1g, ln1b, X2, N);

  // 7) layer-2 GEMM via f32 WMMA: [N,128] x [128,512]
  {
    int rowTiles = (N + 15) / 16;
    int waves = rowTiles * 32;  // 32 col tiles
    gemm_wmma_kernel<128, 128><<<(waves + 7) / 8, TB, 0, stream>>>(X2, W2, H2, N);
  }
  // 8) layer-2 scores
  scores_kernel<<<(N * 4 + TB - 1) / TB, TB, 0, stream>>>(H2, a2, ssrc2, stgt2, N, 128);
  // 9) layer-2 edge softmax
  edge_attn_kernel<<<(E + TB - 1) / TB, TB, 0, stream>>>(eidx, ew, ssrc2, stgt2,
                                                         expe, denom2, E);
  // 10) layer-2 weighted scatter (4 waves per edge, 512 feats)
  {
    long long thr = (long long)E * 128;
    edge_scatter_kernel<128><<<(int)((thr + TB - 1) / TB), TB, 0, stream>>>(
        eidx, expe, denom2, H2, hacc2, E);
  }
  // 11) head-mean + ELU + LayerNorm + residual -> out
  final_kernel<<<(N + 7) / 8, TB, 0, stream>>>(hacc2, X2, ln2g, ln2b, out, N);
}